// MDAM_19129784336962
// MI455X (gfx1250) — compile-verified
//
#include <hip/hip_runtime.h>

typedef __attribute__((ext_vector_type(2))) float v2f;
typedef __attribute__((ext_vector_type(4))) float v4f;
typedef __attribute__((ext_vector_type(8))) float v8f;

#define HW      16384    // 128*128
#define NPLANES 4096     // 256 group-batches * 16 group-channels
#define NB      256      // group-batches (b*GROUP)
#define CG      16       // channels per group
#define KDIM    144      // 16 in-channels * 9 taps

// ---------------------------------------------------------------------------
// Kernel 1: per-(n,c) plane -> 9 shifted-window region sums (zero-pad conv3x3
// mean identity). Needs only: total S, row0/row127 sums, col0/col127 sums,
// 4 corners. Fully coalesced single read of the tensor (regular-temporal:
// x is re-read by k_scale, so let it occupy the 192MB L2).
// ---------------------------------------------------------------------------
__global__ __launch_bounds__(256) void k_reduce(const float* __restrict__ x,
                                                float* __restrict__ R) {
  __shared__ float red[5 * 256];
  __shared__ float corner[4];
  const int t = threadIdx.x;
  const int p = blockIdx.x;                       // plane id = n*16 + c
  const v4f* xv = (const v4f*)(x + (size_t)p * HW);

  float S = 0.f, r0 = 0.f, r127 = 0.f, c0 = 0.f, c127 = 0.f;
  for (int i = t; i < 4096; i += 256) {           // 32 float4 per row
    v4f v = xv[i];
    float s4 = v.x + v.y + v.z + v.w;
    S += s4;
    int row = i >> 5;
    int cs  = (i & 31) << 2;
    if (row == 0)   r0   += s4;
    if (row == 127) r127 += s4;
    if (cs == 0)    c0   += v.x;
    if (cs == 124)  c127 += v.w;
    if (i == 0)    corner[0] = v.x;               // x[0,0]
    if (i == 31)   corner[1] = v.w;               // x[0,127]
    if (i == 4064) corner[2] = v.x;               // x[127,0]
    if (i == 4095) corner[3] = v.w;               // x[127,127]
  }
  red[0*256+t] = S;   red[1*256+t] = r0;  red[2*256+t] = r127;
  red[3*256+t] = c0;  red[4*256+t] = c127;
  __syncthreads();
  for (int s2 = 128; s2 > 0; s2 >>= 1) {
    if (t < s2)
      for (int k = 0; k < 5; ++k) red[k*256+t] += red[k*256+t+s2];
    __syncthreads();
  }
  if (t < 9) {                                    // one region per thread
    float Sa = red[0], R0 = red[256], R127 = red[512], C0 = red[768], C127 = red[1024];
    int dk = t / 3 - 1, dl = t % 3 - 1;           // tap offsets
    float reg = Sa;
    if (dk == -1) reg -= R127; else if (dk == 1) reg -= R0;
    if (dl == -1) reg -= C127; else if (dl == 1) reg -= C0;
    if (dk != 0 && dl != 0)
      reg += corner[(dk == 1 ? 0 : 2) + (dl == 1 ? 0 : 1)];
    int n = p >> 4, c = p & 15;
    // R layout: [K=144][N=256], K index = c*9 + tap; fold 1/HW here
    R[(c * 9 + t) * NB + n] = reg * (1.0f / HW);
  }
}

// ---------------------------------------------------------------------------
// Kernel 2: per-(c,n) attention weights via fp32 WMMA GEMM chain.
//   Ymean(16x256) = W3eff(16x144) @ R(144x256) + b3      (36 wmma steps)
//   Y1 = relu(fc1 @ Ymean)                               (4 wmma steps)
//   Y2 = sigmoid(fc2 @ Y1)                               (4 wmma steps)
//   Wt = sigmoid(softmax(gn_b) + Y2)    [x_att == softmax(gn_b), exact]
// One block = 16 waves; wave w owns N-tile columns [16w, 16w+16).
// Fragment layouts per ISA 7.12.2 (f32, wave32):
//   A 16x4: lanes0-15 M=lane {K0,K1}, lanes16-31 M=lane-16 {K2,K3}
//   B 4x16: lanes0-15 N=lane {K0,K1}, lanes16-31 N=lane-16 {K2,K3}
//   C/D 16x16: vgpr v, lanes0-15 M=v, lanes16-31 M=v+8, N=lane&15
// ---------------------------------------------------------------------------
__global__ __launch_bounds__(512) void k_weights(const float* __restrict__ R,
                                                 const float* __restrict__ w3,
                                                 const float* __restrict__ b3,
                                                 const float* __restrict__ gnb,
                                                 const float* __restrict__ fc1,
                                                 const float* __restrict__ fc2,
                                                 float* __restrict__ Wt) {
  __shared__ float Wlds[CG * KDIM];   // w3 rearranged [o][cin*9+tap] == flat w3
  __shared__ float Ylds[16 * 256];    // per-wave 16x16 staging tiles
  const int t = threadIdx.x;
  const int w = t >> 5, lane = t & 31;
  const int m = lane & 15, hi = lane >> 4;

  for (int i = t; i < CG * KDIM; i += 512) Wlds[i] = w3[i];
  __syncthreads();

  const int col = w * 16 + m;   // n column this lane contributes to

  // GEMM1: 16x144 @ 144x256 (this wave's 16-col tile), K in chunks of 4
  v8f acc = {};
  for (int kk = 0; kk < 36; ++kk) {
    int k0 = kk * 4 + 2 * hi;
    v2f a, b;
    a.x = Wlds[m * KDIM + k0];  a.y = Wlds[m * KDIM + k0 + 1];
    b.x = R[k0 * NB + col];     b.y = R[(k0 + 1) * NB + col];
    acc = __builtin_amdgcn_wmma_f32_16x16x4_f32(false, a, false, b,
                                                (short)0, acc, false, false);
  }
  #pragma unroll
  for (int v = 0; v < 8; ++v) {                 // + b3, stage as [row][col]
    int mr = v + 8 * hi;
    Ylds[w * 256 + mr * 16 + m] = acc[v] + b3[mr];
  }
  __syncthreads();

  // GEMM2: fc1 @ Ymean
  v8f acc2 = {};
  #pragma unroll
  for (int kk = 0; kk < 4; ++kk) {
    int k0 = kk * 4 + 2 * hi;
    v2f a, b;
    a.x = fc1[m * 16 + k0];              a.y = fc1[m * 16 + k0 + 1];
    b.x = Ylds[w * 256 + k0 * 16 + m];   b.y = Ylds[w * 256 + (k0 + 1) * 16 + m];
    acc2 = __builtin_amdgcn_wmma_f32_16x16x4_f32(false, a, false, b,
                                                 (short)0, acc2, false, false);
  }
  __syncthreads();
  #pragma unroll
  for (int v = 0; v < 8; ++v) {                 // relu, restage
    int mr = v + 8 * hi;
    Ylds[w * 256 + mr * 16 + m] = fmaxf(acc2[v], 0.0f);
  }
  __syncthreads();

  // GEMM3: fc2 @ relu(...)
  v8f acc3 = {};
  #pragma unroll
  for (int kk = 0; kk < 4; ++kk) {
    int k0 = kk * 4 + 2 * hi;
    v2f a, b;
    a.x = fc2[m * 16 + k0];              a.y = fc2[m * 16 + k0 + 1];
    b.x = Ylds[w * 256 + k0 * 16 + m];   b.y = Ylds[w * 256 + (k0 + 1) * 16 + m];
    acc3 = __builtin_amdgcn_wmma_f32_16x16x4_f32(false, a, false, b,
                                                 (short)0, acc3, false, false);
  }

  // x_att = softmax(gn_b) over 16 channels (exact identity, see analysis)
  float mx = gnb[0];
  for (int c = 1; c < CG; ++c) mx = fmaxf(mx, gnb[c]);
  float den = 0.f;
  for (int c = 0; c < CG; ++c) den += __expf(gnb[c] - mx);

  #pragma unroll
  for (int v = 0; v < 8; ++v) {
    int mr = v + 8 * hi;
    float y2  = 1.0f / (1.0f + __expf(-acc3[v]));
    float att = __expf(gnb[mr] - mx) / den;
    Wt[mr * NB + col] = 1.0f / (1.0f + __expf(-(att + y2)));
  }
}

// ---------------------------------------------------------------------------
// Kernel 3: out = x * weight[plane]  (pure bandwidth, 128b in/out).
// Non-temporal hints: the 268MB output stream must NOT evict x from the
// 192MB L2 (x was just warmed by k_reduce), and x loads here are last-use.
// ---------------------------------------------------------------------------
__global__ __launch_bounds__(256) void k_scale(const float* __restrict__ x,
                                               const float* __restrict__ Wt,
                                               float* __restrict__ out) {
  const v4f* xv = (const v4f*)x;
  v4f* ov = (v4f*)out;
  const int N4 = NPLANES * (HW / 4);          // 16,777,216 float4s
  const int stride = gridDim.x * blockDim.x;
  for (int i = blockIdx.x * blockDim.x + threadIdx.x; i < N4; i += stride) {
    int plane = i >> 12;                      // 4096 float4 per plane
    float s = Wt[(plane & 15) * NB + (plane >> 4)];
    v4f v = __builtin_nontemporal_load(&xv[i]);   // last-use read of x
    v *= s;
    __builtin_nontemporal_store(v, &ov[i]);       // stream out, bypass L2
  }
}

extern "C" void kernel_launch(void* const* d_in, const int* in_sizes, int n_in,
                              void* d_out, int out_size, void* d_ws, size_t ws_size,
                              hipStream_t stream) {
  // setup_inputs order: inputs, w1, b1, w3, b3, gn_w, gn_b, fc1_w, fc2_w
  const float* x   = (const float*)d_in[0];
  const float* w3  = (const float*)d_in[3];
  const float* b3  = (const float*)d_in[4];
  const float* gnb = (const float*)d_in[6];
  const float* fc1 = (const float*)d_in[7];
  const float* fc2 = (const float*)d_in[8];
  // (w1, b1, gn_w are mathematically dead — see analysis)

  float* R  = (float*)d_ws;           // [144][256]
  float* Wt = R + KDIM * NB;          // [16][256]
  float* out = (float*)d_out;

  k_reduce <<<NPLANES, 256, 0, stream>>>(x, R);
  k_weights<<<1,       512, 0, stream>>>(R, w3, b3, gnb, fc1, fc2, Wt);
  k_scale  <<<16384,   256, 0, stream>>>(x, Wt, out);
}